// TransformerBlockQuantum_65481071397138
// MI455X (gfx1250) — compile-verified
//
#include <hip/hip_runtime.h>
#include <math.h>

typedef __attribute__((ext_vector_type(2))) float v2f;
typedef __attribute__((ext_vector_type(8))) float v8f;

#define E_DIM 1024
#define NQ_DIM 64
#define FFN_DIM 4096
#define NTOK 8192
#define LN_EPS 1e-5f
#define MB 64      // tokens per block (4 M-tiles of 16)
#define KCHUNK 128
#define APAD 132   // 128+4: stride mod 64 banks = 4 -> 16 lanes hit 16 banks
#define QPAD 68    // 64+4

// D = A(16x4, f32) * B(4x16, f32) + C(16x16, f32), full-precision CDNA5 WMMA.
__device__ __forceinline__ v8f wmma_f32(v2f a, v2f b, v8f c) {
  return __builtin_amdgcn_wmma_f32_16x16x4_f32(false, a, false, b, (short)0, c,
                                               false, false);
}

// ---------------------------------------------------------------------------
// Kernel 1: y = x + cos(x + theta_bcast) @ Wc^T
// grid = (NTOK/64)*2 blocks, 256 threads (8 waves).
// Block owns 64 rows x 512 cols; wave w owns cols [nb, +64) (nb = block col base + w*64) as a
// 4(M) x 4(N) tile grid -> each B load feeds 4 WMMAs.
// A is cos-encoded once per block into LDS, in K-chunks of 128.
// ---------------------------------------------------------------------------
__global__ __launch_bounds__(256) void attn_gemm_kernel(
    const float* __restrict__ x, const float* __restrict__ theta,
    const float* __restrict__ Wc, float* __restrict__ y) {
  __shared__ float as[MB][APAD];

  const int wave  = threadIdx.x >> 5;
  const int lane  = threadIdx.x & 31;
  const int laneM = lane & 15;          // A: row M / B: col N within tile
  const int hiK   = (lane >> 4) << 1;   // lane>=16 holds K+2,K+3
  const int mhi   = (lane >> 4) << 3;   // C/D: lane>=16 holds M+8
  const int m0    = (blockIdx.x >> 1) * MB;
  const int nb    = (blockIdx.x & 1) * 512 + wave * 64;

  v8f acc[4][4] = {};

#pragma unroll 1
  for (int kc = 0; kc < E_DIM; kc += KCHUNK) {
    // cooperative load + cos encode of the 64 x 128 A chunk (once per block)
    for (int idx = threadIdx.x; idx < MB * KCHUNK; idx += 256) {
      const int m = idx >> 7, k = idx & (KCHUNK - 1);
      const int kg = kc + k;
      as[m][k] = cosf(x[(size_t)(m0 + m) * E_DIM + kg] + theta[kg & 63]);
    }
    __syncthreads();

#pragma unroll 1
    for (int k0 = 0; k0 < KCHUNK; k0 += 4) {
      const int kg = kc + k0 + hiK;
      v2f b[4], a[4];
#pragma unroll
      for (int t = 0; t < 4; ++t) {  // B[k][n] = Wc[n][k]
        const int n = nb + t * 16 + laneM;
        b[t] = *(const v2f*)(Wc + (size_t)n * E_DIM + kg);
      }
#pragma unroll
      for (int mt = 0; mt < 4; ++mt)
        a[mt] = *(const v2f*)(&as[mt * 16 + laneM][k0 + hiK]);
#pragma unroll
      for (int mt = 0; mt < 4; ++mt)
#pragma unroll
        for (int t = 0; t < 4; ++t)
          acc[mt][t] = wmma_f32(a[mt], b[t], acc[mt][t]);
    }
    __syncthreads();
  }

#pragma unroll
  for (int mt = 0; mt < 4; ++mt)
#pragma unroll
    for (int t = 0; t < 4; ++t) {
      const int n = nb + t * 16 + laneM;
#pragma unroll
      for (int i = 0; i < 8; ++i) {
        const int m = m0 + mt * 16 + mhi + i;
        y[(size_t)m * E_DIM + n] = acc[mt][t][i] + x[(size_t)m * E_DIM + n];
      }
    }
}

// ---------------------------------------------------------------------------
// LayerNorm: one block (256 threads) per token row of 1024.
// ---------------------------------------------------------------------------
__global__ __launch_bounds__(256) void layernorm_kernel(
    const float* __restrict__ y, const float* __restrict__ g,
    const float* __restrict__ b, float* __restrict__ out) {
  __shared__ float red[256];
  const int tid = threadIdx.x;
  const float* row = y + (size_t)blockIdx.x * E_DIM;

  float v[4];
  float s = 0.f;
#pragma unroll
  for (int j = 0; j < 4; ++j) {
    v[j] = row[tid + j * 256];
    s += v[j];
  }
  red[tid] = s;
  __syncthreads();
  for (int off = 128; off > 0; off >>= 1) {
    if (tid < off) red[tid] += red[tid + off];
    __syncthreads();
  }
  const float mu = red[0] * (1.f / E_DIM);
  __syncthreads();

  s = 0.f;
#pragma unroll
  for (int j = 0; j < 4; ++j) {
    const float d = v[j] - mu;
    s += d * d;
  }
  red[tid] = s;
  __syncthreads();
  for (int off = 128; off > 0; off >>= 1) {
    if (tid < off) red[tid] += red[tid + off];
    __syncthreads();
  }
  const float rstd = rsqrtf(red[0] * (1.f / E_DIM) + LN_EPS);

  float* orow = out + (size_t)blockIdx.x * E_DIM;
#pragma unroll
  for (int j = 0; j < 4; ++j) {
    const int c = tid + j * 256;
    orow[c] = (v[j] - mu) * rstd * g[c] + b[c];
  }
}

// ---------------------------------------------------------------------------
// Kernel 2: y2 = x1 + relu(cos(x1[:, :64]+thf) @ W1^T) @ W2^T
// grid = (NTOK/64)*2 blocks, 256 threads. Block owns 64 rows x 512 out cols.
// h (64 x 4096 per block) lives only in LDS, f-chunked by 128.
// Wave owns a 4(M) x 4(N) output tile grid; each W2 load feeds 4 WMMAs.
// ---------------------------------------------------------------------------
__global__ __launch_bounds__(256) void ffn_gemm_kernel(
    const float* __restrict__ x1, const float* __restrict__ thf,
    const float* __restrict__ W1, const float* __restrict__ W2,
    float* __restrict__ y2) {
  __shared__ float qs[MB][QPAD];    // cos-encoded q tile (64 x 64)
  __shared__ float hs[MB][APAD];    // current relu(h) chunk (64 x 128)

  const int wave  = threadIdx.x >> 5;
  const int lane  = threadIdx.x & 31;
  const int laneM = lane & 15;
  const int hiK   = (lane >> 4) << 1;
  const int mhi   = (lane >> 4) << 3;
  const int m0    = (blockIdx.x >> 1) * MB;
  const int eb    = (blockIdx.x & 1) * 512 + wave * 64;

  for (int idx = threadIdx.x; idx < MB * NQ_DIM; idx += 256) {
    const int m = idx >> 6, k = idx & 63;
    qs[m][k] = cosf(x1[(size_t)(m0 + m) * E_DIM + k] + thf[k]);
  }
  __syncthreads();

  v8f acc[4][4] = {};  // 64 x 64 output sub-tile per wave

#pragma unroll 1
  for (int c = 0; c < FFN_DIM / KCHUNK; ++c) {
    const int fbase = c * KCHUNK;

    // --- stage A: hs = relu(q @ W1_chunk^T) ---
    // wave w computes cols [w*16, w*16+16) of the 128-chunk, for 4 M-tiles
    v8f hacc[4] = {};
#pragma unroll 1
    for (int k0 = 0; k0 < NQ_DIM; k0 += 4) {
      const int k = k0 + hiK;
      const int f = fbase + wave * 16 + laneM;  // B[k][f] = W1[f][k]
      v2f b = *(const v2f*)(W1 + (size_t)f * NQ_DIM + k);
#pragma unroll
      for (int mt = 0; mt < 4; ++mt) {
        v2f a = *(const v2f*)(&qs[mt * 16 + laneM][k]);
        hacc[mt] = wmma_f32(a, b, hacc[mt]);
      }
    }
#pragma unroll
    for (int mt = 0; mt < 4; ++mt)
#pragma unroll
      for (int i = 0; i < 8; ++i)
        hs[mt * 16 + mhi + i][wave * 16 + laneM] = fmaxf(hacc[mt][i], 0.f);
    __syncthreads();

    // --- stage B: acc += hs @ W2_chunk^T ---
#pragma unroll 1
    for (int k0 = 0; k0 < KCHUNK; k0 += 4) {
      const int k = k0 + hiK;
      v2f b[4], a[4];
#pragma unroll
      for (int t = 0; t < 4; ++t) {  // B[k][e] = W2[e][fbase+k]
        const int e = eb + t * 16 + laneM;
        b[t] = *(const v2f*)(W2 + (size_t)e * FFN_DIM + fbase + k);
      }
#pragma unroll
      for (int mt = 0; mt < 4; ++mt)
        a[mt] = *(const v2f*)(&hs[mt * 16 + laneM][k]);
#pragma unroll
      for (int mt = 0; mt < 4; ++mt)
#pragma unroll
        for (int t = 0; t < 4; ++t)
          acc[mt][t] = wmma_f32(a[mt], b[t], acc[mt][t]);
    }
    __syncthreads();  // hs reused next chunk
  }

#pragma unroll
  for (int mt = 0; mt < 4; ++mt)
#pragma unroll
    for (int t = 0; t < 4; ++t) {
      const int n = eb + t * 16 + laneM;
#pragma unroll
      for (int i = 0; i < 8; ++i) {
        const int m = m0 + mt * 16 + mhi + i;
        y2[(size_t)m * E_DIM + n] = acc[mt][t][i] + x1[(size_t)m * E_DIM + n];
      }
    }
}

// ---------------------------------------------------------------------------
extern "C" void kernel_launch(void* const* d_in, const int* in_sizes, int n_in,
                              void* d_out, int out_size, void* d_ws,
                              size_t ws_size, hipStream_t stream) {
  const float* x       = (const float*)d_in[0];
  const float* th_attn = (const float*)d_in[1];
  const float* Wc      = (const float*)d_in[2];
  const float* g1      = (const float*)d_in[3];
  const float* b1      = (const float*)d_in[4];
  const float* th_ffn  = (const float*)d_in[5];
  const float* W1      = (const float*)d_in[6];
  const float* W2      = (const float*)d_in[7];
  const float* g2      = (const float*)d_in[8];
  const float* b2      = (const float*)d_in[9];
  float* out = (float*)d_out;

  float* y  = (float*)d_ws;              // NTOK*E floats (reused pre-LN buf)
  float* x1 = y + (size_t)NTOK * E_DIM;  // NTOK*E floats

  dim3 blk(256);
  const int gemm_blocks = (NTOK / MB) * 2;  // 256
  attn_gemm_kernel<<<gemm_blocks, blk, 0, stream>>>(x, th_attn, Wc, y);
  layernorm_kernel<<<NTOK, blk, 0, stream>>>(y, g1, b1, x1);
  ffn_gemm_kernel<<<gemm_blocks, blk, 0, stream>>>(x1, th_ffn, W1, W2, y);
  layernorm_kernel<<<NTOK, blk, 0, stream>>>(y, g2, b2, out);
}